// AFM_layer_22565758173356
// MI455X (gfx1250) — compile-verified
//
#include <hip/hip_runtime.h>
#include <hip/hip_bf16.h>

#define N_DENSE  13
#define N_SPARSE 26
#define EMBED    16
#define VOCAB    100000
#define NCOL     (N_DENSE + N_SPARSE)   // 39

typedef __attribute__((ext_vector_type(2))) float v2f;
typedef __attribute__((ext_vector_type(4))) float v4f;
typedef __attribute__((ext_vector_type(8))) float v8f;

// One wave32 per 16 batch rows.
// Lane L: m = L&15 (row in tile), h = L>>4 (dim half: dims 8h..8h+7).
// Phase 1: gather 26 embedding rows, accumulate sum & sum-of-squares per dim.
// Phase 2: pooled = 0.5*(s^2 - q)  (FM identity == sum over all 325 pairs).
// Phase 3: stage pooled tile in LDS, feed v_wmma_f32_16x16x4_f32 x4 (K=16)
//          against B = [wout | 0 ... 0] to get pooled @ wout in D column 0.
__global__ __launch_bounds__(32) void afm_fused_kernel(
    const int*   __restrict__ inputs,
    const float* __restrict__ tables,
    const float* __restrict__ wout,
    const float* __restrict__ bout,
    float*       __restrict__ out,
    int nrows)
{
  __shared__ float tile[16][17];   // +1 pad to dodge bank conflicts

  const int lane = threadIdx.x;    // 0..31
  const int m    = lane & 15;      // row within 16-row tile / B,D column index
  const int h    = lane >> 4;      // which 8-dim half this lane accumulates

  int row = blockIdx.x * 16 + m;
  if (row >= nrows) row = nrows - 1;   // clamp: keep EXEC all-1s for WMMA

  // Hoist wout pairs (per-chunk B values) + bias so these loads issue early,
  // unconditionally, and overlap the gather stream. Chunk k needs
  // wout[4k+2h], wout[4k+2h+1] -> one aligned 8B load per chunk.
  v2f wk[4];
#pragma unroll
  for (int k = 0; k < 4; ++k)
    wk[k] = *(const v2f*)(wout + 4 * k + 2 * h);
  const float bo = bout[0];

  // ---- Phase 1: gather + accumulate (latency-bound part; 26x2 b128 loads) ----
  v4f s0 = {0.f,0.f,0.f,0.f}, s1 = {0.f,0.f,0.f,0.f};
  v4f q0 = {0.f,0.f,0.f,0.f}, q1 = {0.f,0.f,0.f,0.f};
  const int* idxp = inputs + (size_t)row * NCOL + N_DENSE;
#pragma unroll
  for (int f = 0; f < N_SPARSE; ++f) {
    const int idx = idxp[f];
    const float* e = tables + ((size_t)f * VOCAB + (size_t)idx) * EMBED + 8 * h;
    v4f a = *(const v4f*)(e);
    v4f b = *(const v4f*)(e + 4);
    s0 += a; q0 += a * a;
    s1 += b; q1 += b * b;
  }

  // ---- Phase 2: FM identity: sum over all 325 pairs of elementwise products ----
  v4f p0 = 0.5f * (s0 * s0 - q0);
  v4f p1 = 0.5f * (s1 * s1 - q1);
#pragma unroll
  for (int i = 0; i < 4; ++i) tile[m][8 * h + i]     = p0[i];
#pragma unroll
  for (int i = 0; i < 4; ++i) tile[m][8 * h + 4 + i] = p1[i];
  __syncthreads();

  // ---- Phase 3: D(16x16) = pooled(16x16) x W(16x16), W col0 = wout ----
  // A 16x4 f32 layout: lanes 0-15 {VGPR0:K=4k+0, VGPR1:K=4k+1},
  //                    lanes 16-31 {K=4k+2, K=4k+3}  -> per-lane dims 4k+2h, 4k+2h+1.
  // B 4x16 f32 layout mirrors A's K split; N = lane&15, only N==0 nonzero.
  const bool col0 = (m == 0);
  v8f acc = {};
#pragma unroll
  for (int k = 0; k < 4; ++k) {
    v2f a;
    a.x = tile[m][4 * k + 2 * h];
    a.y = tile[m][4 * k + 2 * h + 1];
    v2f b;                             // register cndmask, no branches
    b.x = col0 ? wk[k].x : 0.0f;
    b.y = col0 ? wk[k].y : 0.0f;
    acc = __builtin_amdgcn_wmma_f32_16x16x4_f32(false, a, false, b,
                                                (short)0, acc, false, false);
  }

  // D col 0: lane 0 holds rows 0..7 (VGPR j -> M=j), lane 16 holds rows 8..15.
  // Each writing lane owns 8 contiguous outputs -> sigmoid in registers, then
  // two 32B-aligned b128 stores (scalar fallback only for a ragged last tile).
  if (col0) {
    const int base = blockIdx.x * 16 + 8 * h;
    float sg[8];
#pragma unroll
    for (int j = 0; j < 8; ++j) {
      const float x = acc[j] + bo;
      sg[j] = __builtin_amdgcn_rcpf(1.0f + __expf(-x));  // v_rcp_f32 sigmoid
    }
    if (base + 8 <= nrows) {
      v4f lo = {sg[0], sg[1], sg[2], sg[3]};
      v4f hi = {sg[4], sg[5], sg[6], sg[7]};
      *(v4f*)(out + base)     = lo;
      *(v4f*)(out + base + 4) = hi;
    } else {
#pragma unroll
      for (int j = 0; j < 8; ++j)
        if (base + j < nrows) out[base + j] = sg[j];
    }
  }
}

extern "C" void kernel_launch(void* const* d_in, const int* in_sizes, int n_in,
                              void* d_out, int out_size, void* d_ws, size_t ws_size,
                              hipStream_t stream) {
  // setup_inputs order: 0 inputs(i32), 1 tables(f32), 2 w1, 3 b1, 4 w2, 5 b2,
  //                     6 wout(f32[16]), 7 bout(f32[1]).
  // w1/b1/w2/b2 are mathematically dead (softmax over a size-1 axis == 1).
  const int*   inputs = (const int*)  d_in[0];
  const float* tables = (const float*)d_in[1];
  const float* wout   = (const float*)d_in[6];
  const float* bout   = (const float*)d_in[7];
  float*       out    = (float*)d_out;

  const int nrows  = in_sizes[0] / NCOL;          // 2048
  const int blocks = (nrows + 15) / 16;           // 128 wave32 workgroups

  afm_fused_kernel<<<blocks, 32, 0, stream>>>(inputs, tables, wout, bout, out, nrows);
}